// CapsNet_5832565588547
// MI455X (gfx1250) — compile-verified
//
#include <hip/hip_runtime.h>
#include <math.h>

// ---- problem constants (from reference) ----
#define BATCH 16
#define CAPS_B 32      // B
#define CAPS_C 32      // C
#define NHH 16         // HH (4x4 pose)
#define KKER 3         // K
#define WOUT 6         // w = (14-3)/2+1
#define BKK 288        // B*K*K
#define CWW 1152       // C*w*w
#define HF 14          // feature map H/W after conv2
#define NPIX 196       // 14*14
#define EPSR 1e-10f
#define LN2PI 1.8378770664093453f

typedef float v2f __attribute__((ext_vector_type(2)));
typedef float v4f __attribute__((ext_vector_type(4)));
typedef float v8f __attribute__((ext_vector_type(8)));

// ------------------------------------------------------------------
// conv1: (16,4,64,64) -> (16,32,30,30), 5x5 stride2 pad0, BN eval + ReLU
// ------------------------------------------------------------------
__global__ void k_conv1(const float* __restrict__ x, const float* __restrict__ w,
                        const float* __restrict__ bias, const float* __restrict__ g,
                        const float* __restrict__ bb, float* __restrict__ f1) {
  int idx = blockIdx.x * blockDim.x + threadIdx.x;
  if (idx >= BATCH * 32 * 30 * 30) return;
  int ox = idx % 30; int t = idx / 30;
  int oy = t % 30; t /= 30;
  int oc = t % 32; int nb = t / 32;
  float s = 0.f;
  for (int ic = 0; ic < 4; ++ic)
    for (int ky = 0; ky < 5; ++ky)
      for (int kx = 0; kx < 5; ++kx)
        s += x[((nb * 4 + ic) * 64 + oy * 2 + ky) * 64 + ox * 2 + kx] *
             w[((oc * 4 + ic) * 5 + ky) * 5 + kx];
  s += bias[oc];
  s = s * (g[oc] * rsqrtf(1.001f)) + bb[oc];
  f1[idx] = s > 0.f ? s : 0.f;
}

// ------------------------------------------------------------------
// conv2: (16,32,30,30) -> (16,32,14,14), 5x5 stride2 pad1, BN eval + ReLU
// ------------------------------------------------------------------
__global__ void k_conv2(const float* __restrict__ f1, const float* __restrict__ w,
                        const float* __restrict__ bias, const float* __restrict__ g,
                        const float* __restrict__ bb, float* __restrict__ f2) {
  int idx = blockIdx.x * blockDim.x + threadIdx.x;
  if (idx >= BATCH * 32 * HF * HF) return;
  int ox = idx % HF; int t = idx / HF;
  int oy = t % HF; t /= HF;
  int oc = t % 32; int nb = t / 32;
  float s = 0.f;
  for (int ic = 0; ic < 32; ++ic)
    for (int ky = 0; ky < 5; ++ky) {
      int iy = oy * 2 + ky - 1;
      if (iy < 0 || iy >= 30) continue;
      for (int kx = 0; kx < 5; ++kx) {
        int ix = ox * 2 + kx - 1;
        if (ix < 0 || ix >= 30) continue;
        s += f1[((nb * 32 + ic) * 30 + iy) * 30 + ix] *
             w[((oc * 32 + ic) * 5 + ky) * 5 + kx];
      }
    }
  s += bias[oc];
  s = s * (g[oc] * rsqrtf(1.001f)) + bb[oc];
  f2[idx] = s > 0.f ? s : 0.f;
}

// ------------------------------------------------------------------
// prim_pose GEMM via WMMA f32 16x16x4: out[o,n] = sum_k W[o,k]*f2[b,k,n]
// stores P[b, Bi, y, x, hh] with o = hh*32 + Bi
// grid (ntile=13, mtile=32, batch=16), block 32 (one wave per 16x16 tile)
// ------------------------------------------------------------------
__global__ void k_gemm_pose(const float* __restrict__ Wm, const float* __restrict__ bias,
                            const float* __restrict__ f2, float* __restrict__ P) {
  int ntile = blockIdx.x, mtile = blockIdx.y, nb = blockIdx.z;
  int lane = threadIdx.x;
  int half = lane >> 4, l = lane & 15;
  int arow = mtile * 16 + l;
  int ncol = ntile * 16 + l;
  v8f acc = {};
  for (int ks = 0; ks < 8; ++ks) {
    int kb = ks * 4 + half * 2;
    v2f a, b;
    a.x = Wm[arow * 32 + kb];
    a.y = Wm[arow * 32 + kb + 1];
    if (ncol < NPIX) {
      b.x = f2[(nb * 32 + kb) * NPIX + ncol];
      b.y = f2[(nb * 32 + kb + 1) * NPIX + ncol];
    } else { b.x = 0.f; b.y = 0.f; }
    acc = __builtin_amdgcn_wmma_f32_16x16x4_f32(false, a, false, b, (short)0, acc,
                                                false, false);
  }
  if (ncol < NPIX) {
    int y = ncol / HF, xx = ncol % HF;
    for (int g2 = 0; g2 < 8; ++g2) {
      int mrow = mtile * 16 + g2 + half * 8;
      float val = acc[g2] + bias[mrow];
      int hh = mrow >> 5, Bi = mrow & 31;
      P[(((nb * 32 + Bi) * HF + y) * HF + xx) * NHH + hh] = val;
    }
  }
}

// prim_act GEMM (M=32) + sigmoid -> acts (b,32,14,14)
__global__ void k_gemm_act(const float* __restrict__ Wm, const float* __restrict__ bias,
                           const float* __restrict__ f2, float* __restrict__ A) {
  int ntile = blockIdx.x, mtile = blockIdx.y, nb = blockIdx.z;
  int lane = threadIdx.x;
  int half = lane >> 4, l = lane & 15;
  int arow = mtile * 16 + l;
  int ncol = ntile * 16 + l;
  v8f acc = {};
  for (int ks = 0; ks < 8; ++ks) {
    int kb = ks * 4 + half * 2;
    v2f a, b;
    a.x = Wm[arow * 32 + kb];
    a.y = Wm[arow * 32 + kb + 1];
    if (ncol < NPIX) {
      b.x = f2[(nb * 32 + kb) * NPIX + ncol];
      b.y = f2[(nb * 32 + kb + 1) * NPIX + ncol];
    } else { b.x = 0.f; b.y = 0.f; }
    acc = __builtin_amdgcn_wmma_f32_16x16x4_f32(false, a, false, b, (short)0, acc,
                                                false, false);
  }
  if (ncol < NPIX) {
    for (int g2 = 0; g2 < 8; ++g2) {
      int mrow = mtile * 16 + g2 + half * 8;
      float val = acc[g2] + bias[mrow];
      val = 1.f / (1.f + expf(-val));
      A[(nb * 32 + mrow) * NPIX + ncol] = val;
    }
  }
}

// ------------------------------------------------------------------
// votes via WMMA: for each (b,i,j,k): [ (c,p) x r ] * [ r x (u,v,q) ]
// M=128, K=4 (one wmma step), N=144.  grid (ntile=9, bkk=288, b=16), block 32.
// V[b, bkk, c*36+u*6+v, p*4+q]   (V stores are non-temporal: write-once stream)
// ------------------------------------------------------------------
__global__ void k_votes(const float* __restrict__ Wc, const float* __restrict__ P,
                        float* __restrict__ V) {
  int ntile = blockIdx.x;
  int bkk = blockIdx.y;
  int nb = blockIdx.z;
  int lane = threadIdx.x;
  int half = lane >> 4, l = lane & 15;
  int iB = bkk / 9, jk = bkk % 9, jj = jk / 3, kk2 = jk % 3;
  int n = ntile * 16 + l;             // < 144
  int u = n / 24, v = (n % 24) >> 2, q = n & 3;
  int row = 2 * v + jj, col = 2 * u + kk2;
  const float* pb = P + (((size_t)(nb * 32 + iB) * HF + row) * HF + col) * NHH;
  v2f bvec;
  bvec.x = pb[(half * 2) * 4 + q];          // h = r*4+q, r = 2*half
  bvec.y = pb[(half * 2 + 1) * 4 + q];
  size_t vbase = ((size_t)(nb * BKK + bkk)) * CWW;
  const float* wb = Wc + (size_t)bkk * 32 * 16;  // (i,j,k) block of C*4*4
  for (int m = 0; m < 8; ++m) {
    int mrow = m * 16 + l;
    int c = mrow >> 2, p = mrow & 3;
    v2f a;
    a.x = wb[c * 16 + p * 4 + half * 2];
    a.y = wb[c * 16 + p * 4 + half * 2 + 1];
    v8f acc = {};
    acc = __builtin_amdgcn_wmma_f32_16x16x4_f32(false, a, false, bvec, (short)0, acc,
                                                false, false);
    for (int g2 = 0; g2 < 8; ++g2) {
      int mr = m * 16 + g2 + half * 8;
      int cc = mr >> 2, pp = mr & 3;
      __builtin_nontemporal_store(acc[g2],
          &V[(vbase + cc * 36 + u * 6 + v) * NHH + pp * 4 + q]);
    }
  }
}

// ------------------------------------------------------------------
// EM routing: stats kernel.  One block per (b,cw); 128 threads = 8 i-slots x 16 h.
// mode 0: R = 1/C (iter 0).  mode 1: R = Rbuf/(den+eps)+eps (lazy normalization).
// Computes sum_R, mu, sigma, log_sigma.  wout: also write mu into d_out.
// V loads are non-temporal (V=340MB > L2, read-once per pass); R/acts stay cached.
// ------------------------------------------------------------------
__global__ void k_stats(const float* __restrict__ V, const float* __restrict__ Rbuf,
                        const float* __restrict__ den, const float* __restrict__ acts,
                        float* __restrict__ muB, float* __restrict__ sigB,
                        float* __restrict__ lsB, float* __restrict__ sumRB,
                        float* __restrict__ out, int mode, int wout) {
  int bid = blockIdx.x;
  int nb = bid / CWW, cw = bid % CWW;
  int uv = cw % 36;
  int u = uv / 6, v = uv % 6;
  int t = threadIdx.x;
  int h = t & 15, isub = t >> 4;
  __shared__ float s_acc[8][16];
  __shared__ float s_ra[8][16];
  __shared__ float s_mu[16];
  __shared__ float s_sumR;

  float mu_num = 0.f, ra_sum = 0.f;
  for (int i = isub; i < BKK; i += 8) {
    int iB = i / 9, jk = i % 9, jj = jk / 3, kk2 = jk % 3;
    float av = acts[(nb * 32 + iB) * NPIX + (2 * v + jj) * HF + (2 * u + kk2)];
    float ra;
    if (mode == 0) ra = av * (1.f / 32.f);
    else {
      float rr = Rbuf[((size_t)(nb * BKK + i)) * CWW + cw];
      ra = (rr / (den[nb * BKK + i] + EPSR) + EPSR) * av;
    }
    float vv = __builtin_nontemporal_load(
        &V[(((size_t)(nb * BKK + i)) * CWW + cw) * NHH + h]);
    mu_num += ra * vv;
    ra_sum += ra;
  }
  s_acc[isub][h] = mu_num;
  s_ra[isub][h] = ra_sum;
  __syncthreads();
  if (t < 16) {
    float tot = 0.f, rt = 0.f;
    for (int s2 = 0; s2 < 8; ++s2) { tot += s_acc[s2][t]; rt += s_ra[s2][t]; }
    float muv = tot / rt;
    s_mu[t] = muv;
    muB[((size_t)(nb * CWW + cw)) * NHH + t] = muv;
    if (wout) out[((size_t)(nb * CWW + cw)) * NHH + t] = muv;
    if (t == 0) { s_sumR = rt; sumRB[nb * CWW + cw] = rt; }
  }
  __syncthreads();
  float muh = s_mu[h];
  float sig_num = 0.f;
  for (int i = isub; i < BKK; i += 8) {
    int iB = i / 9, jk = i % 9, jj = jk / 3, kk2 = jk % 3;
    float av = acts[(nb * 32 + iB) * NPIX + (2 * v + jj) * HF + (2 * u + kk2)];
    float ra;
    if (mode == 0) ra = av * (1.f / 32.f);
    else {
      float rr = Rbuf[((size_t)(nb * BKK + i)) * CWW + cw];
      ra = (rr / (den[nb * BKK + i] + EPSR) + EPSR) * av;
    }
    float vv = __builtin_nontemporal_load(
        &V[(((size_t)(nb * BKK + i)) * CWW + cw) * NHH + h]);
    float d = vv - muh;
    sig_num += ra * d * d;
  }
  s_acc[isub][h] = sig_num;
  __syncthreads();
  if (t < 16) {
    float tot = 0.f;
    for (int s2 = 0; s2 < 8; ++s2) tot += s_acc[s2][t];
    float sg = tot / s_sumR;
    sigB[((size_t)(nb * CWW + cw)) * NHH + t] = sg;
    lsB[((size_t)(nb * CWW + cw)) * NHH + t] = logf(sqrtf(sg) + EPSR);
  }
}

// activation: a = sigmoid(lam*(beta_a[c] - sum_R*(16*beta_v[c] + sum_h log_sigma)))
__global__ void k_act(const float* __restrict__ lsB, const float* __restrict__ sumRB,
                      const float* __restrict__ beta_v, const float* __restrict__ beta_a,
                      const float* __restrict__ lambda_, float* __restrict__ avec,
                      float* __restrict__ out, int wout) {
  int t = blockIdx.x * blockDim.x + threadIdx.x;
  if (t >= BATCH * CWW) return;
  int nb = t / CWW, cw = t % CWW;
  int c = cw / 36;
  const v4f* lp = (const v4f*)(lsB + ((size_t)(nb * CWW + cw)) * NHH);
  float sh = 0.f;
  for (int h4 = 0; h4 < 4; ++h4) {
    v4f lv = lp[h4];
    sh += lv.x + lv.y + lv.z + lv.w;
  }
  float sR = sumRB[nb * CWW + cw];
  float cost = sR * (16.f * beta_v[c] + sh);
  float lam = lambda_[0];
  float av = 1.f / (1.f + expf(-(lam * (beta_a[c] - cost))));
  avec[t] = av;
  if (wout) out[(size_t)BATCH * CWW * NHH + t] = av;
}

// R update: Rbuf <- ap = a * sum_h exp(ln_p); den[b,i] = sum_cw ap (normalized lazily)
// Each thread consumes 16 contiguous floats of V/mu/sig/ls -> B128 vector loads.
__global__ void k_update(const float* __restrict__ V, const float* __restrict__ muB,
                         const float* __restrict__ sigB, const float* __restrict__ lsB,
                         const float* __restrict__ avec, float* __restrict__ Rbuf,
                         float* __restrict__ den) {
  int bid = blockIdx.x;  // nb*288 + i
  int nb = bid / BKK;
  int t = threadIdx.x;
  __shared__ float sd[256];
  float local = 0.f;
  for (int cw = t; cw < CWW; cw += 256) {
    const v4f* vp = (const v4f*)(V + (((size_t)bid) * CWW + cw) * NHH);
    const v4f* mp = (const v4f*)(muB + ((size_t)(nb * CWW + cw)) * NHH);
    const v4f* sp = (const v4f*)(sigB + ((size_t)(nb * CWW + cw)) * NHH);
    const v4f* lp = (const v4f*)(lsB + ((size_t)(nb * CWW + cw)) * NHH);
    float p = 0.f;
    for (int h4 = 0; h4 < 4; ++h4) {
      v4f vv = __builtin_nontemporal_load(vp + h4);   // B128 NT load of V
      v4f mm = mp[h4];
      v4f sg = sp[h4];
      v4f ls = lp[h4];
      for (int e = 0; e < 4; ++e) {
        float d = vv[e] - mm[e];
        p += expf(-d * d / (2.f * sg[e]) - ls[e] - 0.5f * LN2PI);
      }
    }
    float ap = avec[nb * CWW + cw] * p;
    Rbuf[((size_t)bid) * CWW + cw] = ap;
    local += ap;
  }
  sd[t] = local;
  __syncthreads();
  for (int s2 = 128; s2 > 0; s2 >>= 1) {
    if (t < s2) sd[t] += sd[t + s2];
    __syncthreads();
  }
  if (t == 0) den[bid] = sd[0];
}

// ------------------------------------------------------------------
extern "C" void kernel_launch(void* const* d_in, const int* in_sizes, int n_in,
                              void* d_out, int out_size, void* d_ws, size_t ws_size,
                              hipStream_t stream) {
  const float* x          = (const float*)d_in[0];
  const float* lambda_    = (const float*)d_in[1];
  const float* conv1_w    = (const float*)d_in[2];
  const float* conv1_b    = (const float*)d_in[3];
  const float* bn1_g      = (const float*)d_in[4];
  const float* bn1_b      = (const float*)d_in[5];
  const float* conv2_w    = (const float*)d_in[6];
  const float* conv2_b    = (const float*)d_in[7];
  const float* bn2_g      = (const float*)d_in[8];
  const float* bn2_b      = (const float*)d_in[9];
  const float* prim_pose_w= (const float*)d_in[10];
  const float* prim_pose_b= (const float*)d_in[11];
  const float* prim_act_w = (const float*)d_in[12];
  const float* prim_act_b = (const float*)d_in[13];
  const float* W_caps     = (const float*)d_in[14];
  const float* beta_v     = (const float*)d_in[15];
  const float* beta_a     = (const float*)d_in[16];
  float* out = (float*)d_out;

  float* ws = (float*)d_ws;
  size_t o = 0;
  float* f1   = ws + o; o += (size_t)BATCH * 32 * 30 * 30;       // 460800
  float* f2   = ws + o; o += (size_t)BATCH * 32 * HF * HF;       // 100352
  float* P    = ws + o; o += (size_t)BATCH * 32 * HF * HF * NHH; // 1605632
  float* Aact = ws + o; o += (size_t)BATCH * 32 * HF * HF;       // 100352
  float* V    = ws + o; o += (size_t)BATCH * BKK * CWW * NHH;    // 84934656
  float* Rbuf = ws + o; o += (size_t)BATCH * BKK * CWW;          // 5308416
  float* den  = ws + o; o += (size_t)BATCH * BKK;
  float* sumR = ws + o; o += (size_t)BATCH * CWW;
  float* muB  = ws + o; o += (size_t)BATCH * CWW * NHH;
  float* sigB = ws + o; o += (size_t)BATCH * CWW * NHH;
  float* lsB  = ws + o; o += (size_t)BATCH * CWW * NHH;
  float* avec = ws + o; o += (size_t)BATCH * CWW;

  k_conv1<<<1800, 256, 0, stream>>>(x, conv1_w, conv1_b, bn1_g, bn1_b, f1);
  k_conv2<<<392, 256, 0, stream>>>(f1, conv2_w, conv2_b, bn2_g, bn2_b, f2);
  k_gemm_pose<<<dim3(13, 32, BATCH), 32, 0, stream>>>(prim_pose_w, prim_pose_b, f2, P);
  k_gemm_act<<<dim3(13, 2, BATCH), 32, 0, stream>>>(prim_act_w, prim_act_b, f2, Aact);
  k_votes<<<dim3(9, BKK, BATCH), 32, 0, stream>>>(W_caps, P, V);

  for (int it = 0; it < 3; ++it) {
    int mode = (it == 0) ? 0 : 1;
    int wout = (it == 2) ? 1 : 0;
    k_stats<<<BATCH * CWW, 128, 0, stream>>>(V, Rbuf, den, Aact, muB, sigB, lsB,
                                             sumR, out, mode, wout);
    k_act<<<(BATCH * CWW + 255) / 256, 256, 0, stream>>>(lsB, sumR, beta_v, beta_a,
                                                         lambda_, avec, out, wout);
    if (it < 2)
      k_update<<<BATCH * BKK, 256, 0, stream>>>(V, muB, sigB, lsB, avec, Rbuf, den);
  }
}